// E8S_codebook_45990509806222
// MI455X (gfx1250) — compile-verified
//
#include <hip/hip_runtime.h>
#include <hip/hip_bf16.h>

typedef __attribute__((ext_vector_type(2))) float v2f;
typedef __attribute__((ext_vector_type(8))) float v8f;

#define CODESZ 8
#define NEG_BIG -3.0e38f

// One block = 16 rows of X, 8 waves. Each wave strides over 16-codeword tiles,
// doing D = A(16x4) * B(4x16) + C twice (K=8) with C seeded to -0.5*||g||^2.
// The partial last tile (ncw % 16 != 0) is handled in a wave-uniform epilogue
// so the hot loop has no clamps or predicated loads.
__global__ __launch_bounds__(256) void e8_quant_wmma(
    const float* __restrict__ X,     // [nrows, 8]
    const float* __restrict__ G,     // [ncw, 8]
    const float* __restrict__ GN,    // [ncw]
    float* __restrict__ vals_out,    // [nrows, 8]
    short* __restrict__ idx_out,     // [nrows]
    int nrows, int ncw)
{
    const int lane = threadIdx.x & 31;
    // readfirstlane: tell the compiler the wave id is wave-uniform so the loop
    // bound check stays scalar (s_cbranch) and EXEC is never touched around WMMA.
    const int wv   = __builtin_amdgcn_readfirstlane(threadIdx.x >> 5);  // 0..7
    const int c    = lane & 15;            // column-in-tile (B/N) and row-in-tile (A/M)
    const int hi   = lane >> 4;            // which half of the wave
    const int kk   = hi * 2;               // K offset within a K=4 step
    const int row_base = blockIdx.x * 16;

    // --- A tile: loop-invariant. Per ISA layout, lane L holds A[M=L%16][K=kk..kk+1].
    const float* xrow = X + (size_t)(row_base + c) * CODESZ;
    v2f a0, a1;
    a0.x = xrow[kk];     a0.y = xrow[kk + 1];      // K step 0 (k=0..3)
    a1.x = xrow[kk + 4]; a1.y = xrow[kk + 5];      // K step 1 (k=4..7)

    float best[8];
    int   bidx[8];
#pragma unroll
    for (int i = 0; i < 8; ++i) { best[i] = NEG_BIG; bidx[i] = 0; }

    const int nfull = ncw >> 4;            // number of complete 16-codeword tiles
    const int rem   = ncw & 15;

    // --- Hot loop: pointer-increment addressing, no clamps, no branches.
    const float* gb  = G  + (size_t)(wv * 16 + c) * CODESZ + kk;  // B chunk base
    const float* gnp = GN + (wv * 16 + c);                        // norm for this column
    int col = wv * 16 + c;

    for (int t = wv; t < nfull; t += 8) {
        v2f b0 = *(const v2f*)(gb);        // grid[col][kk..kk+1]
        v2f b1 = *(const v2f*)(gb + 4);    // grid[col][kk+4..kk+5]
        const float hn = -0.5f * (*gnp);

        v8f acc = {hn, hn, hn, hn, hn, hn, hn, hn};
        acc = __builtin_amdgcn_wmma_f32_16x16x4_f32(
                  false, a0, false, b0, (short)0, acc, false, false);
        acc = __builtin_amdgcn_wmma_f32_16x16x4_f32(
                  false, a1, false, b1, (short)0, acc, false, false);

#pragma unroll
        for (int i = 0; i < 8; ++i) {
            if (acc[i] > best[i]) { best[i] = acc[i]; bidx[i] = col; }
        }
        gb  += 8 * 16 * CODESZ;            // advance 8 wave-strided tiles
        gnp += 8 * 16;
        col += 8 * 16;
    }

    // --- Tail: the single partial tile, owned by exactly one wave (uniform branch).
    if (rem != 0 && (nfull & 7) == wv) {
        const int tcol = nfull * 16 + c;
        const int colc = (tcol < ncw) ? tcol : (ncw - 1);
        const float* grow = G + (size_t)colc * CODESZ;
        v2f b0, b1;
        b0.x = grow[kk];     b0.y = grow[kk + 1];
        b1.x = grow[kk + 4]; b1.y = grow[kk + 5];
        const float hn = (tcol < ncw) ? (-0.5f * GN[colc]) : NEG_BIG;

        v8f acc = {hn, hn, hn, hn, hn, hn, hn, hn};
        acc = __builtin_amdgcn_wmma_f32_16x16x4_f32(
                  false, a0, false, b0, (short)0, acc, false, false);
        acc = __builtin_amdgcn_wmma_f32_16x16x4_f32(
                  false, a1, false, b1, (short)0, acc, false, false);
#pragma unroll
        for (int i = 0; i < 8; ++i) {
            if (acc[i] > best[i]) { best[i] = acc[i]; bidx[i] = tcol; }
        }
    }

    // --- Cross-lane reduce within each 16-lane half (strides 1,2,4,8 stay in-half).
#pragma unroll
    for (int s = 1; s <= 8; s <<= 1) {
#pragma unroll
        for (int i = 0; i < 8; ++i) {
            float ob = __shfl_xor(best[i], s, 32);
            int   oi = __shfl_xor(bidx[i], s, 32);
            if (ob > best[i] || (ob == best[i] && oi < bidx[i])) {
                best[i] = ob; bidx[i] = oi;
            }
        }
    }

    __shared__ float sbest[8][16];
    __shared__ int   sidx[8][16];
    if ((lane & 15) == 0) {
#pragma unroll
        for (int i = 0; i < 8; ++i) {
            sbest[wv][hi * 8 + i] = best[i];
            sidx [wv][hi * 8 + i] = bidx[i];
        }
    }
    __syncthreads();

    // --- Final reduce across the 8 waves; 16 threads emit 16 rows.
    if (threadIdx.x < 16) {
        const int r = threadIdx.x;
        float b  = sbest[0][r];
        int   bi = sidx[0][r];
#pragma unroll
        for (int w = 1; w < 8; ++w) {
            float ob = sbest[w][r];
            int   oi = sidx [w][r];
            if (ob > b || (ob == b && oi < bi)) { b = ob; bi = oi; }
        }
        const int row = row_base + r;
        const float* grow = G + (size_t)bi * CODESZ;
        float4 g0 = *(const float4*)(grow);
        float4 g1 = *(const float4*)(grow + 4);
        *(float4*)(vals_out + (size_t)row * CODESZ)     = g0;
        *(float4*)(vals_out + (size_t)row * CODESZ + 4) = g1;
        idx_out[row] = (short)(bi - 32768);
    }
}

extern "C" void kernel_launch(void* const* d_in, const int* in_sizes, int n_in,
                              void* d_out, int out_size, void* d_ws, size_t ws_size,
                              hipStream_t stream) {
    const float* X  = (const float*)d_in[0];   // [nrows, 8]
    const float* G  = (const float*)d_in[1];   // [ncw, 8]
    const float* GN = (const float*)d_in[2];   // [ncw]
    const int nrows = in_sizes[0] / CODESZ;    // 8192
    const int ncw   = in_sizes[2];             // 55650

    float* vals_out = (float*)d_out;                               // nrows*8 floats
    short* idx_out  = (short*)(vals_out + (size_t)nrows * CODESZ); // nrows int16 after vals

    const int nblocks = nrows / 16;            // 512
    e8_quant_wmma<<<nblocks, 256, 0, stream>>>(X, G, GN, vals_out, idx_out, nrows, ncw);
}